// Top2Gate_12489764897371
// MI455X (gfx1250) — compile-verified
//
#include <hip/hip_runtime.h>
#include <math.h>

typedef __attribute__((ext_vector_type(2))) float v2f;
typedef __attribute__((ext_vector_type(4))) float v4f;
typedef __attribute__((ext_vector_type(8))) float v8f;

#define S_TOK 8192
#define DIM   2048
#define NEXP  16
#define CAP   1024

// ---------------------------------------------------------------------------
// Pass 0: streaming zero-fill of the ~1.07 GB output with non-temporal b128
// stores (write-once data; keep it out of L2).
// ---------------------------------------------------------------------------
__global__ __launch_bounds__(256) void zero_fill_kernel(float* __restrict__ out,
                                                        size_t n4, size_t total) {
    size_t i = (size_t)blockIdx.x * blockDim.x + threadIdx.x;
    size_t stride = (size_t)gridDim.x * blockDim.x;
    v4f z = {0.0f, 0.0f, 0.0f, 0.0f};
    v4f* o4 = (v4f*)out;
    for (; i < n4; i += stride) {
        __builtin_nontemporal_store(z, o4 + i);
    }
    if (blockIdx.x == 0 && threadIdx.x == 0) {
        for (size_t t = n4 * 4; t < total; ++t) out[t] = 0.0f;
    }
}

// ---------------------------------------------------------------------------
// Pass 1: logits = x @ W.T via V_WMMA_F32_16X16X4_F32.
// One wave per 16-token tile; N = 16 experts exactly fills the WMMA N dim.
// 8 waves/block, 64 blocks -> 512 tiles. EXEC is all-ones (no divergence).
// ---------------------------------------------------------------------------
__global__ __launch_bounds__(256) void gemm_logits_kernel(const float* __restrict__ x,
                                                          const float* __restrict__ W,
                                                          float* __restrict__ logits) {
    const int lane = threadIdx.x & 31;
    const int wave = threadIdx.x >> 5;
    const int tile = blockIdx.x * 8 + wave;          // 0..511
    const int rowBase = tile * 16;
    const int m    = lane & 15;                      // A row / B col / C col
    const int kOff = (lane >> 4) * 2;                // K sub-offset per half-wave

    const float* __restrict__ xa = x + (size_t)(rowBase + m) * DIM + kOff;
    const float* __restrict__ wb = W + (size_t)m * DIM + kOff;   // B = W.T, col-major = W rows

    v8f acc = {};
    for (int k = 0; k < DIM; k += 4) {
        v2f a = *(const v2f*)(xa + k);
        v2f b = *(const v2f*)(wb + k);
        acc = __builtin_amdgcn_wmma_f32_16x16x4_f32(
            /*neg_a=*/false, a, /*neg_b=*/false, b,
            /*c_mod=*/(short)0, acc, /*reuse_a=*/false, /*reuse_b=*/false);
    }

    const int rOff = (lane >> 4) * 8;                // C row-half per lane group
#pragma unroll
    for (int r = 0; r < 8; ++r) {
        logits[(size_t)(rowBase + rOff + r) * NEXP + m] = acc[r];
    }
}

// ---------------------------------------------------------------------------
// Pass 2: per-token softmax over 16 experts, top-1 of gates, top-1 of
// (logits + gumbel noise) with expert-1 masked to -inf.
// ---------------------------------------------------------------------------
__global__ __launch_bounds__(256) void gate_token_kernel(const float* __restrict__ logits,
                                                         const float* __restrict__ noise,
                                                         float* __restrict__ gates,
                                                         int* __restrict__ idx1o,
                                                         int* __restrict__ idx2o,
                                                         float* __restrict__ g1o,
                                                         float* __restrict__ g2o) {
    const int s = blockIdx.x * blockDim.x + threadIdx.x;   // 0..8191
    float lg[NEXP];
#pragma unroll
    for (int e = 0; e < NEXP; ++e) lg[e] = logits[(size_t)s * NEXP + e];

    // softmax (max-subtracted)
    float mx = lg[0];
#pragma unroll
    for (int e = 1; e < NEXP; ++e) mx = fmaxf(mx, lg[e]);
    float ex[NEXP];
    float sum = 0.0f;
#pragma unroll
    for (int e = 0; e < NEXP; ++e) { ex[e] = expf(lg[e] - mx); sum += ex[e]; }
    const float inv = 1.0f / sum;

    // argmax of gates == argmax of logits (softmax is order-preserving, first-max wins)
    int i1 = 0; float b1 = lg[0];
#pragma unroll
    for (int e = 1; e < NEXP; ++e) { if (lg[e] > b1) { b1 = lg[e]; i1 = e; } }

    // second expert: argmax of logits + noise, with i1 masked to -inf
    int i2 = 0; float b2 = -INFINITY;
#pragma unroll
    for (int e = 0; e < NEXP; ++e) {
        float v = (e == i1) ? -INFINITY : (lg[e] + noise[(size_t)s * NEXP + e]);
        if (v > b2) { b2 = v; i2 = e; }
    }

#pragma unroll
    for (int e = 0; e < NEXP; ++e) gates[(size_t)s * NEXP + e] = ex[e] * inv;
    idx1o[s] = i1;
    idx2o[s] = i2;
    g1o[s] = ex[i1] * inv;
    g2o[s] = ex[i2] * inv;
}

// ---------------------------------------------------------------------------
// Pass 3: two-level exclusive scan of one-hot masks over tokens, per expert.
// 256 threads = 16 experts x 16 segments of 512 tokens. Also computes l_aux.
// locations2 gets +sum(mask1, axis=0) per the reference.
// ---------------------------------------------------------------------------
__global__ __launch_bounds__(256) void scan_kernel(const int* __restrict__ idx1,
                                                   const int* __restrict__ idx2,
                                                   const float* __restrict__ gates,
                                                   int* __restrict__ loc1,
                                                   int* __restrict__ loc2,
                                                   float* __restrict__ out /* d_out, writes [0] */) {
    __shared__ int   c1s[256];
    __shared__ int   c2s[256];
    __shared__ float sgs[256];
    __shared__ float partial[NEXP];

    const int t = threadIdx.x;
    const int e = t & 15;
    const int seg = t >> 4;
    const int s0 = seg * (S_TOK / 16);
    const int s1 = s0 + (S_TOK / 16);

    int c1 = 0, c2 = 0; float sg = 0.0f;
    for (int s = s0; s < s1; ++s) {
        c1 += (idx1[s] == e);
        c2 += (idx2[s] == e);
        sg += gates[(size_t)s * NEXP + e];
    }
    c1s[t] = c1; c2s[t] = c2; sgs[t] = sg;
    __syncthreads();

    int off1 = 0, off2 = 0, tot1 = 0;
#pragma unroll
    for (int j = 0; j < 16; ++j) {
        int v1 = c1s[j * 16 + e];
        int v2 = c2s[j * 16 + e];
        tot1 += v1;
        if (j < seg) { off1 += v1; off2 += v2; }
    }
    off2 += tot1;    // locations2 += sum(mask1, axis=0)

    for (int s = s0; s < s1; ++s) {
        if (idx1[s] == e) loc1[s] = off1++;
        if (idx2[s] == e) loc2[s] = off2++;
    }

    if (seg == 0) {
        float me = 0.0f;
#pragma unroll
        for (int j = 0; j < 16; ++j) me += sgs[j * 16 + e];
        me *= (1.0f / (float)S_TOK);
        float ce = (float)tot1 * (1.0f / (float)S_TOK);
        partial[e] = me * ce;
    }
    __syncthreads();
    if (t == 0) {
        float l = 0.0f;
#pragma unroll
        for (int j = 0; j < NEXP; ++j) l += partial[j];
        out[0] = l * (1.0f / (float)NEXP);   // l_aux = mean over E of me*ce
    }
}

// ---------------------------------------------------------------------------
// Pass 4: sparse scatter of the <=2 nonzero (expert, slot) entries per token
// into combine [S,E,C] and dispatch_mask [S,E,C] (as 0/1 float).
// ---------------------------------------------------------------------------
__global__ __launch_bounds__(256) void scatter_kernel(const int* __restrict__ idx1,
                                                      const int* __restrict__ idx2,
                                                      const int* __restrict__ loc1,
                                                      const int* __restrict__ loc2,
                                                      const float* __restrict__ g1a,
                                                      const float* __restrict__ g2a,
                                                      float* __restrict__ out) {
    const int s = blockIdx.x * blockDim.x + threadIdx.x;   // 0..8191
    const int e1 = idx1[s], e2 = idx2[s];
    const int l1 = loc1[s], l2 = loc2[s];
    const bool k1 = (l1 < CAP);
    const bool k2 = (l2 < CAP);
    const float a = k1 ? g1a[s] : 0.0f;
    const float b = k2 ? g2a[s] : 0.0f;
    const float denom = fmaxf(a + b, 1.1920928955078125e-7f);  // FLT_EPSILON
    const float w1 = a / denom;
    const float w2 = b / denom;

    float* __restrict__ combine  = out + 1;
    float* __restrict__ dispatch = out + 1 + (size_t)S_TOK * NEXP * CAP;
    const size_t base = (size_t)s * NEXP * CAP;

    if (k1) {
        const size_t o = base + (size_t)e1 * CAP + l1;
        combine[o]  = w1;
        dispatch[o] = 1.0f;
    }
    if (k2) {
        const size_t o = base + (size_t)e2 * CAP + l2;
        combine[o]  = w2;
        dispatch[o] = 1.0f;
    }
}

// ---------------------------------------------------------------------------
extern "C" void kernel_launch(void* const* d_in, const int* in_sizes, int n_in,
                              void* d_out, int out_size, void* d_ws, size_t ws_size,
                              hipStream_t stream) {
    const float* x     = (const float*)d_in[0];   // [8192, 2048]
    const float* W     = (const float*)d_in[1];   // [16, 2048]
    const float* noise = (const float*)d_in[2];   // [8192, 16]
    float* out = (float*)d_out;

    // workspace carve-out (~1.25 MB)
    char* ws = (char*)d_ws;
    float* logits = (float*)ws;  ws += (size_t)S_TOK * NEXP * sizeof(float);
    float* gates  = (float*)ws;  ws += (size_t)S_TOK * NEXP * sizeof(float);
    int*   idx1   = (int*)ws;    ws += (size_t)S_TOK * sizeof(int);
    int*   idx2   = (int*)ws;    ws += (size_t)S_TOK * sizeof(int);
    int*   loc1   = (int*)ws;    ws += (size_t)S_TOK * sizeof(int);
    int*   loc2   = (int*)ws;    ws += (size_t)S_TOK * sizeof(int);
    float* g1     = (float*)ws;  ws += (size_t)S_TOK * sizeof(float);
    float* g2     = (float*)ws;  ws += (size_t)S_TOK * sizeof(float);

    // 0) zero the whole output (1.07 GB streamed with NT b128 stores)
    size_t total = (size_t)out_size;
    size_t n4 = total / 4;
    zero_fill_kernel<<<4096, 256, 0, stream>>>(out, n4, total);

    // 1) logits = x @ W.T via f32 WMMA; 512 tiles of 16 tokens, 8 waves/block
    gemm_logits_kernel<<<64, 256, 0, stream>>>(x, W, logits);

    // 2) per-token softmax / top-1 / noisy top-2
    gate_token_kernel<<<S_TOK / 256, 256, 0, stream>>>(logits, noise, gates,
                                                       idx1, idx2, g1, g2);

    // 3) per-expert two-level scan over tokens + l_aux -> out[0]
    scan_kernel<<<1, 256, 0, stream>>>(idx1, idx2, gates, loc1, loc2, out);

    // 4) sparse scatter of combine weights and dispatch mask
    scatter_kernel<<<S_TOK / 256, 256, 0, stream>>>(idx1, idx2, loc1, loc2,
                                                    g1, g2, out);
}